// GCNNet_mse_35347580846782
// MI455X (gfx1250) — compile-verified
//
#include <hip/hip_runtime.h>

#define N_NODES 50000
#define N_EDGES 1600000
#define D 128
#define N_LAYERS 6
#define N_GRAPHS 8
#define BN_EPS 1e-5f

typedef __attribute__((ext_vector_type(2))) float v2f;
typedef __attribute__((ext_vector_type(8))) float v8f;
typedef __attribute__((ext_vector_type(4))) unsigned int u32x4;
typedef __attribute__((ext_vector_type(8))) int i32x8;
typedef __attribute__((ext_vector_type(4))) int i32x4;

__device__ __forceinline__ void atomAddG(float* p, float v) {
    (void)__hip_atomic_fetch_add(p, v, __ATOMIC_RELAXED, __HIP_MEMORY_SCOPE_AGENT);
}
__device__ __forceinline__ void atomAddL(float* p, float v) {
    (void)__hip_atomic_fetch_add(p, v, __ATOMIC_RELAXED, __HIP_MEMORY_SCOPE_WORKGROUP);
}

// -------- zero agg buffer (N*D floats) and BN stat accumulators (256 floats) --------
__global__ void k_zero(float* __restrict__ agg, float* __restrict__ stats) {
    int i = blockIdx.x * blockDim.x + threadIdx.x;
    float4 z = {0.f, 0.f, 0.f, 0.f};
    if (i < (N_NODES * D) / 4) ((float4*)agg)[i] = z;
    if (blockIdx.x == 0 && threadIdx.x < 2 * D) stats[threadIdx.x] = 0.f;
}

// -------- edge scatter: agg[dst] += h[src]; one wave per edge, float4 per lane --------
__global__ void k_scatter(const float* __restrict__ h, const int* __restrict__ src,
                          const int* __restrict__ dst, float* __restrict__ agg) {
    int gid = blockIdx.x * blockDim.x + threadIdx.x;
    int e = gid >> 5;
    int lane = gid & 31;
    if (e >= N_EDGES) return;
    int s = src[e];
    int d = dst[e];
    const float4 v = ((const float4*)(h + (size_t)s * D))[lane];
    float* out = agg + (size_t)d * D + lane * 4;
    atomAddG(out + 0, v.x);
    atomAddG(out + 1, v.y);
    atomAddG(out + 2, v.z);
    atomAddG(out + 3, v.w);
}

// Build D# group0 for a TDM 2D tile load (count=1 valid, type=2 "image")
__device__ __forceinline__ u32x4 tdm_group0(unsigned lds_addr, unsigned long long gaddr) {
    u32x4 g;
    g.x = 1u;                                    // count=1, user mode, no gather
    g.y = lds_addr;                              // LDS byte address
    g.z = (unsigned)gaddr;                       // global_addr[31:0]
    g.w = (unsigned)(gaddr >> 32) | (2u << 30);  // global_addr[56:32] | type=2
    return g;
}

// D# group1 for a 16x128 f32 tile: data_size=4B, LDS pad 4 dwords every 128 dwords
__device__ __forceinline__ i32x8 tdm_group1_tile16x128() {
    i32x8 g;
    g.s0 = (int)((2u << 16)      // data_size = 4 bytes
               | (1u << 20)      // pad_enable
               | (6u << 22)      // pad_interval code 6 = every 128 dwords
               | (3u << 25));    // pad_amount code 3 = 4 dwords (128 -> 132 row stride)
    g.s1 = (int)(128u << 16);    // tensor_dim0 = 128 (bits 79:48, low half)
    g.s2 = (int)(16u << 16);     // tensor_dim0 hi=0 | tensor_dim1 = 16 (bits 111:80)
    g.s3 = (int)(128u << 16);    // tensor_dim1 hi=0 | tile_dim0 = 128 (bits 127:112)
    g.s4 = 16;                   // tile_dim1 = 16, tile_dim2 = 0
    g.s5 = 128;                  // tensor_dim0_stride = 128 (bits 207:160, low)
    g.s6 = 0;                    // stride hi | tensor_dim1_stride lo
    g.s7 = 0;
    return g;
}

// -------- fused: hout = relu(agg@Wrel + hin@Wroot + brel); accumulate BN sum/sumsq ----
// One block per 16-row tile (3125 blocks, exact => EXEC all-ones for WMMA),
// 8 waves = 8 column tiles of 16. A-tiles staged via Tensor Data Mover (TDM).
__global__ __launch_bounds__(256) void k_gemm_bn(
    const float* __restrict__ agg, const float* __restrict__ hin,
    const float* __restrict__ Wrel, const float* __restrict__ Wroot,
    const float* __restrict__ brel, float* __restrict__ hout,
    float* __restrict__ gsum, float* __restrict__ gsq) {
    __shared__ float As[16][132];   // 132-dword row stride via TDM pad (banks spread)
    __shared__ float Hs[16][132];
    __shared__ float sSum[D];
    __shared__ float sSq[D];

    const int t = threadIdx.x;
    const int m0 = blockIdx.x * 16;
    if (t < D) { sSum[t] = 0.f; sSq[t] = 0.f; }
    if (t == 0) { __builtin_prefetch(Wrel, 0, 1); __builtin_prefetch(Wroot, 0, 1); }

    // Wave 0 issues two async TDM tile loads (global -> LDS, padded rows),
    // waits TENSORcnt, then the block barrier releases the consumer waves.
    if (t < 32) {
        unsigned ldsA = (unsigned)(size_t)(&As[0][0]);   // low 32 bits = LDS offset
        unsigned ldsH = (unsigned)(size_t)(&Hs[0][0]);
        unsigned long long gA = (unsigned long long)(size_t)(agg + (size_t)m0 * D);
        unsigned long long gH = (unsigned long long)(size_t)(hin + (size_t)m0 * D);
        i32x8 g1 = tdm_group1_tile16x128();
        i32x4 gz4 = {0, 0, 0, 0};
        i32x8 gz8 = {0, 0, 0, 0, 0, 0, 0, 0};
        __builtin_amdgcn_tensor_load_to_lds(tdm_group0(ldsA, gA), g1, gz4, gz4, gz8, 0);
        __builtin_amdgcn_tensor_load_to_lds(tdm_group0(ldsH, gH), g1, gz4, gz4, gz8, 0);
        __builtin_amdgcn_s_wait_tensorcnt(0);
    }
    __syncthreads();

    const int lane = t & 31;
    const int wave = t >> 5;
    const int hlf  = lane >> 4;            // 0: K pair {0,1}, 1: K pair {2,3}
    const int l16  = lane & 15;
    const int n0   = wave * 16;
    const int j    = n0 + l16;             // output column
    const int koff = hlf * 2;

    // Two independent accumulator chains (agg@Wrel and hin@Wroot) for 2x WMMA ILP.
    v8f c1 = {0.f, 0.f, 0.f, 0.f, 0.f, 0.f, 0.f, 0.f};
    v8f c2 = {0.f, 0.f, 0.f, 0.f, 0.f, 0.f, 0.f, 0.f};
    #pragma unroll
    for (int k0 = 0; k0 < D; k0 += 4) {
        v2f a1; a1.x = As[l16][k0 + koff]; a1.y = As[l16][k0 + koff + 1];
        v2f b1; b1.x = Wrel[(size_t)(k0 + koff) * D + j];
                b1.y = Wrel[(size_t)(k0 + koff + 1) * D + j];
        c1 = __builtin_amdgcn_wmma_f32_16x16x4_f32(false, a1, false, b1, (short)0, c1, false, false);
        v2f a2; a2.x = Hs[l16][k0 + koff]; a2.y = Hs[l16][k0 + koff + 1];
        v2f b2; b2.x = Wroot[(size_t)(k0 + koff) * D + j];
                b2.y = Wroot[(size_t)(k0 + koff + 1) * D + j];
        c2 = __builtin_amdgcn_wmma_f32_16x16x4_f32(false, a2, false, b2, (short)0, c2, false, false);
    }

    const float bias = brel[j];
    float s = 0.f, ss = 0.f;
    #pragma unroll
    for (int v = 0; v < 8; ++v) {
        float val = c1[v] + c2[v] + bias;  // C/D layout: VGPR v, half hlf -> row m0+v+8*hlf
        val = val > 0.f ? val : 0.f;
        hout[(size_t)(m0 + v + hlf * 8) * D + j] = val;
        s += val;
        ss += val * val;
    }
    atomAddL(&sSum[j], s);
    atomAddL(&sSq[j], ss);
    __syncthreads();
    if (t < D) { atomAddG(&gsum[t], sSum[t]); atomAddG(&gsq[t], sSq[t]); }
}

// -------- BN: scale/shift from accumulated stats (biased variance, training mode) ----
__global__ void k_bn_finalize(const float* __restrict__ gsum, const float* __restrict__ gsq,
                              const float* __restrict__ gamma, const float* __restrict__ beta,
                              float* __restrict__ scale, float* __restrict__ shift) {
    int j = threadIdx.x;
    if (j < D) {
        const float inv_n = 1.0f / (float)N_NODES;
        float mean = gsum[j] * inv_n;
        float var  = gsq[j] * inv_n - mean * mean;
        float r    = rsqrtf(var + BN_EPS);
        float sc   = gamma[j] * r;
        scale[j] = sc;
        shift[j] = beta[j] - mean * sc;
    }
}

__global__ void k_bn_apply(float* __restrict__ h, const float* __restrict__ scale,
                           const float* __restrict__ shift) {
    int i = blockIdx.x * blockDim.x + threadIdx.x;    // over N*D/4 float4s
    if (i >= (N_NODES * D) / 4) return;
    int j4 = (i & 31) * 4;                            // D/4 == 32 quads per row
    float4 v = ((float4*)h)[i];
    v.x = v.x * scale[j4 + 0] + shift[j4 + 0];
    v.y = v.y * scale[j4 + 1] + shift[j4 + 1];
    v.z = v.z * scale[j4 + 2] + shift[j4 + 2];
    v.w = v.w * scale[j4 + 3] + shift[j4 + 3];
    ((float4*)h)[i] = v;
}

// -------- global mean pool + dense head --------
__global__ void k_zero_pool(float* __restrict__ pooled, float* __restrict__ counts) {
    int i = blockIdx.x * blockDim.x + threadIdx.x;
    if (i < N_GRAPHS * D) pooled[i] = 0.f;
    if (i < N_GRAPHS) counts[i] = 0.f;
}

__global__ void k_pool(const float* __restrict__ h, const int* __restrict__ batch,
                       float* __restrict__ pooled, float* __restrict__ counts) {
    int gid = blockIdx.x * blockDim.x + threadIdx.x;  // N_NODES*D threads
    int node = gid >> 7;
    int f = gid & 127;
    if (node >= N_NODES) return;
    int g = batch[node];
    atomAddG(&pooled[g * D + f], h[(size_t)node * D + f]);
    if (f == 0) atomAddG(&counts[g], 1.0f);
}

__global__ void k_head(const float* __restrict__ pooled, const float* __restrict__ counts,
                       const float* __restrict__ dw, const float* __restrict__ db,
                       const float* __restrict__ mw, const float* __restrict__ mb,
                       float* __restrict__ out) {
    __shared__ float pm[D];
    __shared__ float red[D];
    int t = threadIdx.x;
    for (int g = 0; g < N_GRAPHS; ++g) {
        float cnt = counts[g];
        cnt = cnt > 1.f ? cnt : 1.f;
        pm[t] = pooled[g * D + t] / cnt;
        __syncthreads();
        float acc = db[t];
        for (int k = 0; k < D; ++k) acc += pm[k] * dw[k * D + t];
        acc = acc > 0.f ? acc : 0.f;                  // relu
        red[t] = acc * mw[t];
        __syncthreads();
        for (int off = 64; off > 0; off >>= 1) {
            if (t < off) red[t] += red[t + off];
            __syncthreads();
        }
        if (t == 0) out[g] = red[0] + mb[0];
        __syncthreads();
    }
}

extern "C" void kernel_launch(void* const* d_in, const int* in_sizes, int n_in,
                              void* d_out, int out_size, void* d_ws, size_t ws_size,
                              hipStream_t stream) {
    const float* x      = (const float*)d_in[0];
    const int*   edge   = (const int*)d_in[1];     // [2, N_EDGES]
    const int*   batch  = (const int*)d_in[2];     // [N_NODES]
    const float* Wrel   = (const float*)d_in[3];   // [L, D, D]
    const float* brel   = (const float*)d_in[4];   // [L, D]
    const float* Wroot  = (const float*)d_in[5];   // [L, D, D]
    const float* gamma  = (const float*)d_in[6];   // [L, D]
    const float* beta   = (const float*)d_in[7];   // [L, D]
    const float* dw     = (const float*)d_in[8];   // [D, D]
    const float* db     = (const float*)d_in[9];   // [D]
    const float* mw     = (const float*)d_in[10];  // [D, 1]
    const float* mb     = (const float*)d_in[11];  // [1]
    float* out = (float*)d_out;

    char* ws = (char*)d_ws;
    const size_t MAT = (size_t)N_NODES * D * sizeof(float);   // 25.6 MB
    float* agg   = (float*)(ws);
    float* bufA  = (float*)(ws + MAT);
    float* bufB  = (float*)(ws + 2 * MAT);
    float* stats = (float*)(ws + 3 * MAT);
    float* gsum   = stats;
    float* gsq    = stats + D;
    float* scale  = stats + 2 * D;
    float* shift  = stats + 3 * D;
    float* pooled = stats + 4 * D;
    float* counts = pooled + N_GRAPHS * D;

    const int* src = edge;
    const int* dst = edge + N_EDGES;

    const float* hin = x;
    float* hout = bufA;
    for (int i = 0; i < N_LAYERS; ++i) {
        k_zero<<<(N_NODES * D / 4 + 255) / 256, 256, 0, stream>>>(agg, gsum);
        k_scatter<<<(N_EDGES * 32) / 256, 256, 0, stream>>>(hin, src, dst, agg);
        k_gemm_bn<<<N_NODES / 16, 256, 0, stream>>>(
            agg, hin, Wrel + (size_t)i * D * D, Wroot + (size_t)i * D * D,
            brel + i * D, hout, gsum, gsq);
        k_bn_finalize<<<1, 128, 0, stream>>>(gsum, gsq, gamma + i * D, beta + i * D, scale, shift);
        k_bn_apply<<<(N_NODES * D / 4 + 255) / 256, 256, 0, stream>>>(hout, scale, shift);
        hin = hout;
        hout = (hout == bufA) ? bufB : bufA;
    }
    k_zero_pool<<<4, 256, 0, stream>>>(pooled, counts);
    k_pool<<<(N_NODES * D) / 256, 256, 0, stream>>>(hin, batch, pooled, counts);
    k_head<<<1, 128, 0, stream>>>(pooled, counts, dw, db, mw, mb, out);
}